// MultiResolutionHashEncoding_12446815224198
// MI455X (gfx1250) — compile-verified
//
#include <hip/hip_runtime.h>
#include <hip/hip_bf16.h>

// Multi-resolution hash encoding (Instant-NGP style, nearest-cell / no interp).
//
// Roofline: 24 MB coord reads + 256 MB output writes + random 8B gathers into
// 64 MB of tables. MI455X L2 = 192 MB -> all 16 level tables stay L2-resident
// as long as the streaming traffic doesn't evict them:
//   - output stores are non-temporal (TH=NT)
//   - coord loads are non-temporal (read once)
//   - table gathers use default RT policy -> L2 resident
// Hash math is exact in uint32 because TABLE_SIZE = 2^19 and h >= 0:
//   h % 2^19 == h & 0x7FFFF, and the low 19 bits of int64 mul/xor equal the
//   low 19 bits of the same computation in wrap-around uint32.

#define HE_TBITS 19u
#define HE_TABLE_SIZE (1u << HE_TBITS)
#define HE_N_LEVELS 16u
#define HE_COORDS_DIM 3u

typedef __attribute__((ext_vector_type(2))) float v2f;

// Stored as float so no per-thread u32->f32 conversion is needed; all values
// are small integers, exactly representable, matching jnp.float32(resolution).
__constant__ float kResolutionsF[HE_N_LEVELS] = {
    16.f, 23.f, 32.f, 45.f, 64.f, 91.f, 128.f, 181.f,
    256.f, 362.f, 512.f, 724.f, 1024.f, 1448.f, 2048.f, 2896.f};

__global__ __launch_bounds__(256) void MultiResolutionHashEncoding_12446815224198_kernel(
    const float* __restrict__ coords,   // (B, 3) float32
    const float* __restrict__ tables,   // (16, 2^19, 2) float32
    float* __restrict__ out,            // (B, 32) float32
    unsigned n_points)
{
    // One thread per (point, level). Lanes 0..15 of a point share its coords
    // (single broadcast b96 fetch) and write 16 consecutive float2's
    // -> one contiguous, fully coalesced 128B output line per point.
    const unsigned tid   = blockIdx.x * blockDim.x + threadIdx.x;
    const unsigned level = tid & (HE_N_LEVELS - 1u);
    const unsigned point = tid >> 4;
    if (point >= n_points) return;

    const float* c = coords + (size_t)point * HE_COORDS_DIM;
    const float x = __builtin_nontemporal_load(c + 0);
    const float y = __builtin_nontemporal_load(c + 1);
    const float z = __builtin_nontemporal_load(c + 2);

    const float res = kResolutionsF[level];

    // floor() then int-convert: coords in [0,1) so values are non-negative;
    // matches jnp.floor(coords * float32(res)).astype(int64) bit-exactly.
    const unsigned gx = (unsigned)(int)floorf(x * res);
    const unsigned gy = (unsigned)(int)floorf(y * res);
    const unsigned gz = (unsigned)(int)floorf(z * res);

    // Spatial hash in wrap-around uint32 (low 19 bits identical to int64 ref).
    const unsigned h   = gx ^ (gy * 2654435761u) ^ (gz * 805459861u);
    const unsigned idx = h & (HE_TABLE_SIZE - 1u);

    // 8-byte gather from the level's table. Tables total 64 MB -> L2 resident.
    const v2f* __restrict__ tab = reinterpret_cast<const v2f*>(tables);
    const v2f feat = tab[((size_t)level << HE_TBITS) + idx];

    // Non-temporal store: keep the 256 MB output stream out of L2 so the
    // hash tables stay resident.
    v2f* o = reinterpret_cast<v2f*>(out) + ((size_t)point * HE_N_LEVELS + level);
    __builtin_nontemporal_store(feat, o);
}

extern "C" void kernel_launch(void* const* d_in, const int* in_sizes, int n_in,
                              void* d_out, int out_size, void* d_ws, size_t ws_size,
                              hipStream_t stream) {
    (void)n_in; (void)out_size; (void)d_ws; (void)ws_size;

    const float* coords = (const float*)d_in[0];  // (B, 3)
    const float* tables = (const float*)d_in[1];  // (16, 2^19, 2)
    float* out = (float*)d_out;                   // (B, 32)

    const unsigned n_points = (unsigned)(in_sizes[0] / (int)HE_COORDS_DIM);
    const unsigned total_threads = n_points * HE_N_LEVELS;  // 33.5M -> fits u32
    const unsigned block = 256u;
    const unsigned grid = (total_threads + block - 1u) / block;

    MultiResolutionHashEncoding_12446815224198_kernel<<<grid, block, 0, stream>>>(
        coords, tables, out, n_points);
}